// BayesianAtlas_12567074308525
// MI455X (gfx1250) — compile-verified
//
#include <hip/hip_runtime.h>

// Problem constants (match reference setup_inputs exactly)
#define BATCH 16
#define NPTS  131072
#define GRD   32
#define G3    (GRD * GRD * GRD)
#define TPB   256

// ---- gfx1250 async global->LDS copy (ASYNCcnt path), GV addressing mode ----
__device__ __forceinline__ void async_dw_to_lds(unsigned lds_byte_addr, const float* gaddr) {
  // copies one dword from gaddr to LDS byte offset lds_byte_addr;  tracked by ASYNCcnt
  asm volatile("global_load_async_to_lds_b32 %0, %1, off"
               :: "v"(lds_byte_addr), "v"(gaddr)
               : "memory");
}
__device__ __forceinline__ void wait_asynccnt0() {
  asm volatile("s_wait_asynccnt 0x0" ::: "memory");
}

// ---- Pre-pass: B x 3 x G^3 (component-planar) -> B x G^3 x float4 (interleaved) ----
// Makes each trilinear corner a single 16B global_load_b128 from L2 instead of
// three 4B loads 128KB apart. Source field is read exactly once -> NT loads;
// packed field is the hot reuse structure -> default (RT) store.
__global__ __launch_bounds__(TPB) void pack_field_kernel(const float* __restrict__ vel,
                                                         float4* __restrict__ field) {
  int idx  = blockIdx.x * TPB + threadIdx.x;   // [0, BATCH*G3)
  int b    = idx >> 15;                        // /G3
  int cell = idx & (G3 - 1);
  const float* p = vel + (size_t)b * 3 * G3 + cell;
  float vx = __builtin_nontemporal_load(p);
  float vy = __builtin_nontemporal_load(p + G3);
  float vz = __builtin_nontemporal_load(p + 2 * G3);
  field[idx] = make_float4(vx, vy, vz, 0.0f);
}

// ---- Main flow kernel: T-1 Euler steps, x kept in registers ----
template <bool PACKED>
__global__ __launch_bounds__(TPB) void flow_kernel(const float4* __restrict__ field,
                                                   const float* __restrict__ vel,
                                                   const float* __restrict__ points,
                                                   const float* __restrict__ bbox,
                                                   const int* __restrict__ ntp,
                                                   float* __restrict__ out) {
  __shared__ float spts[TPB * 3];

  const int b  = blockIdx.y;
  const int t  = threadIdx.x;
  const int pt = blockIdx.x * TPB + t;         // NPTS % TPB == 0

  // Stage this block's point tile into LDS via the async path (ASYNCcnt).
  const float* gp = points + ((size_t)b * NPTS + pt) * 3;
  unsigned lds = (unsigned)(uintptr_t)(&spts[t * 3]);   // low 32 bits of flat addr = LDS offset
  async_dw_to_lds(lds + 0, gp + 0);
  async_dw_to_lds(lds + 4, gp + 1);
  async_dw_to_lds(lds + 8, gp + 2);

  // Uniform scalars (overlap with async copy in flight).
  const float l0 = bbox[0], h0 = bbox[1];
  const float l1 = bbox[2], h1 = bbox[3];
  const float l2 = bbox[4], h2 = bbox[5];
  const float s0 = (float)(GRD - 1) / (h0 - l0);
  const float s1 = (float)(GRD - 1) / (h1 - l1);
  const float s2 = (float)(GRD - 1) / (h2 - l2);
  const int   steps = ntp[0] - 1;              // T - 1
  const float dt    = 1.0f / (float)steps;

  wait_asynccnt0();
  float x = spts[t * 3 + 0];
  float y = spts[t * 3 + 1];
  float z = spts[t * 3 + 2];

  const float4* __restrict__ fb = PACKED ? (field + (size_t)b * G3) : nullptr;
  const float*  __restrict__ vb = PACKED ? nullptr : (vel + (size_t)b * 3 * G3);

  for (int it = 0; it < steps; ++it) {
    // Grid coordinates
    float u = (x - l0) * s0;
    float v = (y - l1) * s1;
    float w = (z - l2) * s2;

    // lo = clip(floor(u), 0, G-1); hi = clip(lo+1, 0, G-1)
    float lu = fminf(fmaxf(floorf(u), 0.0f), (float)(GRD - 1));
    float lv = fminf(fmaxf(floorf(v), 0.0f), (float)(GRD - 1));
    float lw = fminf(fmaxf(floorf(w), 0.0f), (float)(GRD - 1));
    int iu0 = (int)lu, iv0 = (int)lv, iw0 = (int)lw;
    int iu1 = min(iu0 + 1, GRD - 1);
    int iv1 = min(iv0 + 1, GRD - 1);
    int iw1 = min(iw0 + 1, GRD - 1);

    // f = uvw - lo ; g = lo + 1 - uvw   (exactly as reference)
    float fu = u - lu, fv = v - lv, fw = w - lw;
    float gu = (lu + 1.0f) - u, gv = (lv + 1.0f) - v, gw = (lw + 1.0f) - w;

    int r0 = iu0 << 10, r1 = iu1 << 10;   // iu * G^2
    int c0 = iv0 << 5,  c1 = iv1 << 5;    // iv * G

    float ax = 0.0f, ay = 0.0f, az = 0.0f;

    // Same corner order as reference: (cu, cv, cw) = 000,001,010,011,100,101,110,111
    if (PACKED) {
      float4 cn;
      #define ACC(R, C, IW, W)                                   \
        { cn = fb[(R) + (C) + (IW)]; float wt = (W);             \
          ax = fmaf(wt, cn.x, ax); ay = fmaf(wt, cn.y, ay);      \
          az = fmaf(wt, cn.z, az); }
      ACC(r0, c0, iw0, gu * gv * gw)
      ACC(r0, c0, iw1, gu * gv * fw)
      ACC(r0, c1, iw0, gu * fv * gw)
      ACC(r0, c1, iw1, gu * fv * fw)
      ACC(r1, c0, iw0, fu * gv * gw)
      ACC(r1, c0, iw1, fu * gv * fw)
      ACC(r1, c1, iw0, fu * fv * gw)
      ACC(r1, c1, iw1, fu * fv * fw)
      #undef ACC
    } else {
      #define ACC(R, C, IW, W)                                   \
        { int ii = (R) + (C) + (IW); float wt = (W);             \
          ax = fmaf(wt, vb[ii], ax);                             \
          ay = fmaf(wt, vb[ii + G3], ay);                        \
          az = fmaf(wt, vb[ii + 2 * G3], az); }
      ACC(r0, c0, iw0, gu * gv * gw)
      ACC(r0, c0, iw1, gu * gv * fw)
      ACC(r0, c1, iw0, gu * fv * gw)
      ACC(r0, c1, iw1, gu * fv * fw)
      ACC(r1, c0, iw0, fu * gv * gw)
      ACC(r1, c0, iw1, fu * gv * fw)
      ACC(r1, c1, iw0, fu * fv * gw)
      ACC(r1, c1, iw1, fu * fv * fw)
      #undef ACC
    }

    // Euler step
    x = fmaf(dt, ax, x);
    y = fmaf(dt, ay, y);
    z = fmaf(dt, az, z);
  }

  // Write-once output: non-temporal so the packed field stays resident in L2.
  float* op = out + ((size_t)b * NPTS + pt) * 3;
  __builtin_nontemporal_store(x, op + 0);
  __builtin_nontemporal_store(y, op + 1);
  __builtin_nontemporal_store(z, op + 2);
}

extern "C" void kernel_launch(void* const* d_in, const int* in_sizes, int n_in,
                              void* d_out, int out_size, void* d_ws, size_t ws_size,
                              hipStream_t stream) {
  (void)in_sizes; (void)n_in; (void)out_size;

  const float* vel  = (const float*)d_in[0];   // B x 3 x G x G x G, f32
  const float* pts  = (const float*)d_in[1];   // B x N x 3, f32
  const float* bbox = (const float*)d_in[2];   // 3 x 2, f32
  const int*   ntp  = (const int*)d_in[3];     // scalar T
  float* out = (float*)d_out;                  // B x N x 3, f32

  dim3 grid(NPTS / TPB, BATCH);
  const size_t packed_bytes = (size_t)BATCH * G3 * sizeof(float4);

  if (ws_size >= packed_bytes) {
    float4* field = (float4*)d_ws;
    pack_field_kernel<<<(BATCH * G3) / TPB, TPB, 0, stream>>>(vel, field);
    flow_kernel<true><<<grid, TPB, 0, stream>>>(field, vel, pts, bbox, ntp, out);
  } else {
    flow_kernel<false><<<grid, TPB, 0, stream>>>(nullptr, vel, pts, bbox, ntp, out);
  }
}